// second_depooling_48636209660361
// MI455X (gfx1250) — compile-verified
//
#include <hip/hip_runtime.h>

typedef float v2f __attribute__((ext_vector_type(2)));
typedef float v8f __attribute__((ext_vector_type(8)));

#define KPAD 52    // K = 49 padded to multiple of 4
#define PPAD 176   // P = 169 padded to multiple of 16 (11 tiles)

// Constant linear map, built on device once per launch. Stored transposed:
// g_Mt[k * PPAD + p] == M[p][k], zero-padded.
__device__ float g_Mt[KPAD * PPAD];

__global__ void build_M_kernel() {
  __shared__ int base_src[169];
  for (int i = threadIdx.x; i < KPAD * PPAD; i += blockDim.x) g_Mt[i] = 0.0f;
  __syncthreads();
  if (threadIdx.x != 0) return;

  const int BASE[33][2] = {
    {1,0},{3,0},{5,0},{7,0},{9,0},{11,0},{0,2},{2,2},{4,2},{6,2},{8,2},
    {10,2},{12,2},{1,4},{3,4},{5,4},{7,4},{9,4},{11,4},{2,6},{4,6},{6,6},
    {8,6},{10,6},{3,8},{5,8},{7,8},{9,8},{4,10},{6,10},{8,10},{5,12},{7,12}};
  const int UNEVEN[52][2] = {
    {5,1},{6,1},{7,1},{3,1},{0,1},{4,1},{9,1},{2,1},{10,1},{1,1},{11,1},
    {8,1},{6,3},{3,3},{7,3},{4,3},{8,3},{2,3},{9,3},{1,3},{10,3},{0,3},
    {11,3},{5,3},{6,5},{4,5},{10,5},{1,5},{9,5},{5,5},{2,5},{8,5},{7,5},
    {3,5},{4,7},{6,7},{9,7},{5,7},{8,7},{3,7},{7,7},{2,7},{6,9},{5,9},
    {7,9},{8,9},{3,9},{4,9},{4,11},{7,11},{5,11},{6,11}};
  const int EVEN[26][2] = {
    {4,0},{6,0},{10,0},{2,0},{8,0},{5,2},{7,2},{3,2},{9,2},{1,2},{11,2},
    {2,4},{8,4},{10,4},{6,4},{4,4},{7,6},{9,6},{5,6},{3,6},{4,8},{6,8},
    {8,8},{5,10},{7,10},{6,12}};

  for (int i = 0; i < 169; ++i) base_src[i] = -1;

  // BASE: direct copy, weight 1. MAP = BASE//2 ; row += (col//2 + 1) % 2
  for (int n = 0; n < 33; ++n) {
    int br = BASE[n][0], bc = BASE[n][1];
    int mc = bc >> 1;
    int mr = (br >> 1) + ((mc & 1) ^ 1);
    int src = mr * 7 + mc;
    base_src[br * 13 + bc] = src;
    g_Mt[src * PPAD + (br * 13 + bc)] = 1.0f;
  }
  // UNEVEN: average of the BASE neighbors (structural nonzero count).
  for (int n = 0; n < 52; ++n) {
    int i = UNEVEN[n][0], j = UNEVEN[n][1];
    int jm = (j - 1 < 0) ? 0 : j - 1;
    int jp = (j + 1 > 12) ? 12 : j + 1;
    int ip = (i + 1 > 12) ? 12 : i + 1;
    int nb[4] = { i * 13 + jm, i * 13 + jp, ip * 13 + jm, ip * 13 + jp };
    int cnt = 0;
    for (int q = 0; q < 4; ++q)
      if (base_src[nb[q]] >= 0) cnt++;
    float wgt = 1.0f / (float)(cnt > 1 ? cnt : 1);
    int p = i * 13 + j;
    for (int q = 0; q < 4; ++q) {
      int s = base_src[nb[q]];
      if (s >= 0) g_Mt[s * PPAD + p] += wgt;
    }
  }
  // EVEN: mean of two vertically adjacent input pixels (cnt2 == 2 a.s.).
  for (int n = 0; n < 26; ++n) {
    int er = EVEN[n][0], ec = EVEN[n][1];
    int hr = er >> 1, hc = ec >> 1;
    int r0 = hr < 6 ? hr : 6;
    int r1 = (hr + 1) < 6 ? (hr + 1) : 6;
    int p = er * 13 + ec;
    g_Mt[(r0 * 7 + hc) * PPAD + p] += 0.5f;
    g_Mt[(r1 * 7 + hc) * PPAD + p] += 0.5f;
  }
}

// GEMM: out(N x 169) = X(N x 49) * Mt(49 x 169) via v_wmma_f32_16x16x4_f32.
// Block = 352 threads = 11 wave32s; block handles 16 images, wave w handles
// output columns [16w, 16w+16).
__global__ __launch_bounds__(352) void depool_wmma_kernel(
    const float* __restrict__ x, float* __restrict__ out) {
  __shared__ float sx[16 * KPAD];

  const int tid = threadIdx.x;
  const long imgBase = (long)blockIdx.x << 4;
  const float* xblk = x + imgBase * 49;

  // Coalesced load of 16 images * 49 floats, de-interleaved into padded LDS.
  for (int t = tid; t < 16 * 49; t += 352) {
    int row = t / 49;
    int k = t - row * 49;
    sx[row * KPAD + k] = xblk[t];
  }
  if (tid < 48) {  // zero K padding (k = 49..51 per row)
    int row = tid / 3;
    int k = 49 + (tid - row * 3);
    sx[row * KPAD + k] = 0.0f;
  }
  __syncthreads();

  const int wv   = tid >> 5;       // wave id: column tile 0..10
  const int lane = tid & 31;
  const int half = lane >> 4;      // 0: K pair {0,1}; 1: K pair {2,3}
  const int mrow = lane & 15;      // matrix row (A) / column-in-tile (B,D)
  const int p    = (wv << 4) + mrow;

  v8f acc = {};
#pragma unroll
  for (int kk = 0; kk < 13; ++kk) {
    const int k0 = kk * 4 + half * 2;
    v2f a, b;
    // A 16x4 layout: lanes 0-15 -> K=k0,k0+1 ; lanes 16-31 -> K=k0+2,k0+3
    a.x = sx[mrow * KPAD + k0];
    a.y = sx[mrow * KPAD + k0 + 1];
    // B 4x16 layout: per-VGPR a K-row striped across lanes (N = lane & 15)
    b.x = g_Mt[k0 * PPAD + p];
    b.y = g_Mt[(k0 + 1) * PPAD + p];
    acc = __builtin_amdgcn_wmma_f32_16x16x4_f32(
        /*neg_a=*/false, a, /*neg_b=*/false, b,
        /*c_mod=*/(short)0, acc, /*reuse_a=*/false, /*reuse_b=*/false);
  }

  // D 16x16 f32 layout: VGPR g, lanes 0-15 -> M=g ; lanes 16-31 -> M=8+g
  if (p < 169) {
    const long obase = imgBase * 169 + p;
#pragma unroll
    for (int g = 0; g < 8; ++g) {
      int r = g + (half << 3);
      out[obase + (long)r * 169] = acc[g];
    }
  }
}

extern "C" void kernel_launch(void* const* d_in, const int* in_sizes, int n_in,
                              void* d_out, int out_size, void* d_ws, size_t ws_size,
                              hipStream_t stream) {
  (void)n_in; (void)out_size; (void)d_ws; (void)ws_size;
  const float* x = (const float*)d_in[0];
  float* out = (float*)d_out;

  const int nImages = in_sizes[0] / 49;   // 256 * 2048 = 524288
  const int nBlocks = nImages >> 4;       // 16 images per block

  build_M_kernel<<<1, 256, 0, stream>>>();
  depool_wmma_kernel<<<nBlocks, 352, 0, stream>>>(x, out);
}